// CoordRegressionNetwork_Fushion_75290776699449
// MI455X (gfx1250) — compile-verified
//
#include <hip/hip_runtime.h>
#include <hip/hip_bf16.h>
#include <math.h>

// Problem constants (from reference)
#define BB   16
#define CIN  512
#define CC   64
#define HH   64
#define WW   160
#define HWN  (HH * WW)          // 10240

typedef float v2f __attribute__((ext_vector_type(2)));
typedef float v8f __attribute__((ext_vector_type(8)));
typedef unsigned int v4u __attribute__((ext_vector_type(4)));
typedef int v4i __attribute__((ext_vector_type(4)));
typedef int v8i __attribute__((ext_vector_type(8)));

__device__ __forceinline__ v8f wmma_f32(v2f a, v2f b, v8f c) {
  // V_WMMA_F32_16X16X4_F32 : D(16x16,f32) = A(16x4,f32) x B(4x16,f32) + C
  return __builtin_amdgcn_wmma_f32_16x16x4_f32(
      /*neg_a=*/false, a, /*neg_b=*/false, b,
      /*c_mod=*/(short)0, c, /*reuse_a=*/false, /*reuse_b=*/false);
}

#define V8F_ZERO {0.f,0.f,0.f,0.f,0.f,0.f,0.f,0.f}

#if __has_builtin(__builtin_amdgcn_tensor_load_to_lds) && \
    __has_builtin(__builtin_amdgcn_s_wait_tensorcnt)
#define USE_TDM 1
#else
#define USE_TDM 0
#endif

#if USE_TDM
// Tensor Data Mover: load a [64 rows x 128 cols] f32 tile of hm_w
// (tensor_dim0 = 512, row stride 512) into LDS with +2 DWORD padding every
// 128 DWORDs -> LDS row stride 130 floats (bank spreading for WMMA A-frags).
__device__ __forceinline__ void tdm_load_hmw_chunk(const float* gsrc,
                                                   unsigned lds_off) {
  const unsigned long long ga = (unsigned long long)(const void*)gsrc;
  v4u g0;
  g0[0] = 1u;                                   // count=1 (valid), user mode
  g0[1] = lds_off;                              // lds_addr (bytes)
  g0[2] = (unsigned)(ga & 0xffffffffu);         // global_addr[31:0]
  g0[3] = (unsigned)((ga >> 32) & 0x01ffffffu)  // global_addr[56:32]
          | (2u << 30);                         // type = 2 ("image")
  v8i g1;
  g1[0] = (int)((2u << 16)    // data_size = 2 -> 4 bytes
              | (1u << 20)    // pad_enable
              | (6u << 22)    // pad_interval = 6 -> every 128 DWORDs
              | (1u << 25));  // pad_amount  = 1 -> 2 DWORDs
  g1[1] = (int)(512u << 16);  // tensor_dim0 = 512 (low16 at bits 63:48)
  g1[2] = (int)(64u << 16);   // tensor_dim0 hi = 0 | tensor_dim1 = 64
  g1[3] = (int)(128u << 16);  // tensor_dim1 hi = 0 | tile_dim0 = 128
  g1[4] = 64;                 // tile_dim1 = 64, tile_dim2 = 0
  g1[5] = 512;                // tensor_dim0_stride = 512
  g1[6] = 0;
  g1[7] = 0;
  const v4i z4 = {0, 0, 0, 0};
#if __has_include(<hip/amd_detail/amd_gfx1250_TDM.h>)
  const v8i z8 = {0, 0, 0, 0, 0, 0, 0, 0};
  __builtin_amdgcn_tensor_load_to_lds(g0, g1, z4, z4, z8, 0);   // clang-23 form
#else
  __builtin_amdgcn_tensor_load_to_lds(g0, g1, z4, z4, 0);       // ROCm 7.2 form
#endif
}
#endif  // USE_TDM

// ---------------------------------------------------------------------------
// Kernel 1: unnorm[b,c,hw] = sum_i hm_w[c,i] * fcn[b,i,hw]   (1x1 conv GEMM)
// Grid: (HWN/128, BB), block 256 (8 waves). Wave owns one 16-col N-tile and
// computes all 64 output rows (4 M-tiles) so fcn is read exactly once.
// ---------------------------------------------------------------------------
__global__ __launch_bounds__(256)
void k_gemm1(const float* __restrict__ fcn, const float* __restrict__ hm_w,
             float* __restrict__ unnorm) {
  const int b    = blockIdx.y;
  const int tid  = threadIdx.x;
  const int lane = tid & 31;
  const int wave = tid >> 5;
  const int col  = blockIdx.x * 128 + wave * 16 + (lane & 15);
  const int krow = (lane >> 4) << 1;            // 0 or 2 (A/B K sub-row)

  __shared__ float sA[64 * 130];                // hm_w chunk [64][128] pad+2

  const float* fcn_b = fcn + (size_t)b * CIN * HWN;

  v8f acc[4] = {V8F_ZERO, V8F_ZERO, V8F_ZERO, V8F_ZERO};

  for (int k0 = 0; k0 < CIN; k0 += 128) {
#if USE_TDM
    // Stage hm_w[:, k0:k0+128] via the Tensor Data Mover (one DMA per chunk,
    // issued by wave 0; TDM ignores EXEC, tracked with TENSORcnt).
    if (wave == 0) {
      tdm_load_hmw_chunk(hm_w + k0, (unsigned)(size_t)sA);
      __builtin_amdgcn_s_wait_tensorcnt(0);
    }
#else
    for (int t = tid; t < 64 * 32; t += 256) {
      const int row = t >> 5;
      const int c4  = (t & 31) << 2;
      const float4 v = *(const float4*)(hm_w + row * CIN + k0 + c4);
      float* d = &sA[row * 130 + c4];
      d[0] = v.x; d[1] = v.y; d[2] = v.z; d[3] = v.w;
    }
#endif
    __syncthreads();

    if (k0 + 128 < CIN)  // hint next fcn chunk into cache
      __builtin_prefetch(fcn_b + (size_t)(k0 + 128 + krow) * HWN + col, 0, 0);

#pragma unroll 4
    for (int kk = 0; kk < 128; kk += 4) {
      const float* bp = fcn_b + (size_t)(k0 + kk + krow) * HWN + col;
      v2f bf; bf.x = bp[0]; bf.y = bp[HWN];
#pragma unroll
      for (int t = 0; t < 4; ++t) {
        const v2f af = *(const v2f*)&sA[(t * 16 + (lane & 15)) * 130 + kk + krow];
        acc[t] = wmma_f32(af, bf, acc[t]);
      }
    }
    __syncthreads();
  }

  const int rbase = 8 * (lane >> 4);
  float* ob = unnorm + (size_t)b * CC * HWN + col;
#pragma unroll
  for (int t = 0; t < 4; ++t)
#pragma unroll
    for (int r = 0; r < 8; ++r)
      ob[(size_t)(t * 16 + rbase + r) * HWN] = acc[t][r];
}

// ---------------------------------------------------------------------------
// Softmax over HW (one block per (b,c) row) + DSNT coordinate expectation.
// ---------------------------------------------------------------------------
__global__ __launch_bounds__(256)
void k_softmax_dsnt(const float* __restrict__ in, float* __restrict__ outp,
                    float* __restrict__ coords) {
  const int row = blockIdx.x;                  // b*CC + c
  const int tid = threadIdx.x;
  const float* x = in + (size_t)row * HWN;
  float* o = outp + (size_t)row * HWN;
  __shared__ float red[256];

  float m = -3.4e38f;
  for (int i = tid; i < HWN; i += 256) m = fmaxf(m, x[i]);
  red[tid] = m; __syncthreads();
  for (int s = 128; s > 0; s >>= 1) {
    if (tid < s) red[tid] = fmaxf(red[tid], red[tid + s]);
    __syncthreads();
  }
  m = red[0]; __syncthreads();

  float sum = 0.f;
  for (int i = tid; i < HWN; i += 256) sum += expf(x[i] - m);
  red[tid] = sum; __syncthreads();
  for (int s = 128; s > 0; s >>= 1) {
    if (tid < s) red[tid] += red[tid + s];
    __syncthreads();
  }
  const float inv = 1.f / red[0]; __syncthreads();

  float sx = 0.f, sy = 0.f;
  for (int i = tid; i < HWN; i += 256) {
    const float p = expf(x[i] - m) * inv;
    o[i] = p;
    const int w = i % WW, h = i / WW;
    sx += p * ((2.f * w + 1.f) / WW - 1.f);
    sy += p * ((2.f * h + 1.f) / HH - 1.f);
  }
  red[tid] = sx; __syncthreads();
  for (int s = 128; s > 0; s >>= 1) {
    if (tid < s) red[tid] += red[tid + s];
    __syncthreads();
  }
  sx = red[0]; __syncthreads();
  red[tid] = sy; __syncthreads();
  for (int s = 128; s > 0; s >>= 1) {
    if (tid < s) red[tid] += red[tid + s];
    __syncthreads();
  }
  if (tid == 0) { coords[row * 2 + 0] = sx; coords[row * 2 + 1] = red[0]; }
}

// ---------------------------------------------------------------------------
// Depthwise 5x5 conv, SAME padding, groups = CC.  dw_w layout [C][1][5][5].
// ---------------------------------------------------------------------------
__global__ __launch_bounds__(256)
void k_dwconv(const float* __restrict__ in, const float* __restrict__ wgt,
              float* __restrict__ out) {
  const int idx = blockIdx.x * 256 + threadIdx.x;   // < BB*CC*HH*WW
  const int w = idx % WW;
  const int h = (idx / WW) % HH;
  const int bc = idx / HWN;                          // b*CC + c
  const int c = bc % CC;
  const float* ib = in + (size_t)bc * HWN;
  const float* wk = wgt + c * 25;
  float acc = 0.f;
#pragma unroll
  for (int ky = 0; ky < 5; ++ky) {
    const int hy = h + ky - 2;
    if (hy < 0 || hy >= HH) continue;
#pragma unroll
    for (int kx = 0; kx < 5; ++kx) {
      const int wx = w + kx - 2;
      if (wx < 0 || wx >= WW) continue;
      acc += ib[hy * WW + wx] * wk[ky * 5 + kx];
    }
  }
  out[idx] = acc;
}

// ---------------------------------------------------------------------------
__global__ __launch_bounds__(256)
void k_zero(float* __restrict__ p, int n) {
  const int i = blockIdx.x * 256 + threadIdx.x;
  if (i < n) p[i] = 0.f;
}

// ---------------------------------------------------------------------------
// relation[b,c,d] = sum_n feat[b,c,n] * feat[b,d,n].
// Grid (16 tiles, BB, 8 K-slices), block 32 (1 wave per 16x16 tile/slice).
// Split-K partials combined with GLOBAL_ATOMIC_ADD_F32 into zeroed rel.
// ---------------------------------------------------------------------------
__global__ __launch_bounds__(32)
void k_relation(const float* __restrict__ feat, float* __restrict__ rel) {
  const int b    = blockIdx.y;
  const int mi   = blockIdx.x >> 2;
  const int ni   = blockIdx.x & 3;
  const int k0   = blockIdx.z * (HWN / 8);
  const int lane = threadIdx.x & 31;
  const int krow = (lane >> 4) << 1;
  const float* base = feat + (size_t)b * CC * HWN;
  const float* ap = base + (size_t)(16 * mi + (lane & 15)) * HWN + krow + k0;
  const float* bp = base + (size_t)(16 * ni + (lane & 15)) * HWN + krow + k0;

  v8f acc = V8F_ZERO;
#pragma unroll 8
  for (int kk = 0; kk < HWN / 8; kk += 4) {
    const v2f af = *(const v2f*)(ap + kk);
    const v2f bf = *(const v2f*)(bp + kk);
    acc = wmma_f32(af, bf, acc);
  }

  const int rowb = 16 * mi + 8 * (lane >> 4);
  const int colo = 16 * ni + (lane & 15);
  float* ob = rel + (size_t)b * CC * CC;
#pragma unroll
  for (int r = 0; r < 8; ++r)
    atomicAdd(&ob[(rowb + r) * CC + colo], acc[r]);
}

// ---------------------------------------------------------------------------
// Row softmax of (rowmax - relation) over last dim (64). In-place.
// ---------------------------------------------------------------------------
__global__ __launch_bounds__(64)
void k_rel_softmax(float* __restrict__ rel) {
  const int row = blockIdx.x;                  // b*CC + c
  const int tid = threadIdx.x;
  __shared__ float red[64];
  const float v = rel[row * 64 + tid];
  red[tid] = v; __syncthreads();
  for (int s = 32; s > 0; s >>= 1) {
    if (tid < s) red[tid] = fmaxf(red[tid], red[tid + s]);
    __syncthreads();
  }
  const float rowmax = red[0]; __syncthreads();
  const float rn = rowmax - v;
  red[tid] = rn; __syncthreads();
  for (int s = 32; s > 0; s >>= 1) {
    if (tid < s) red[tid] = fmaxf(red[tid], red[tid + s]);
    __syncthreads();
  }
  const float m2 = red[0]; __syncthreads();
  const float e = expf(rn - m2);
  red[tid] = e; __syncthreads();
  for (int s = 32; s > 0; s >>= 1) {
    if (tid < s) red[tid] += red[tid + s];
    __syncthreads();
  }
  rel[row * 64 + tid] = e / red[0];
}

// ---------------------------------------------------------------------------
// pre2[b,c,n] = (sum_d R[b,c,d]*feat[b,d,n] + feat[b,c,n]) * fu_w[c]
// Grid (HWN/128, BB), block 256; R staged in LDS (padded stride 66).
// ---------------------------------------------------------------------------
__global__ __launch_bounds__(256)
void k_feat_e(const float* __restrict__ R, const float* __restrict__ feat,
              const float* __restrict__ fu, float* __restrict__ outp) {
  const int b    = blockIdx.y;
  const int tid  = threadIdx.x;
  const int lane = tid & 31;
  const int wave = tid >> 5;
  const int col  = blockIdx.x * 128 + wave * 16 + (lane & 15);
  const int krow = (lane >> 4) << 1;

  __shared__ float sR[64 * 66];
  for (int t = tid; t < 64 * 64; t += 256)
    sR[(t >> 6) * 66 + (t & 63)] = R[(size_t)b * CC * CC + t];
  __syncthreads();

  const float* hb = feat + (size_t)b * CC * HWN;
  v8f acc[4] = {V8F_ZERO, V8F_ZERO, V8F_ZERO, V8F_ZERO};

#pragma unroll
  for (int kk = 0; kk < 64; kk += 4) {
    const float* bp = hb + (size_t)(kk + krow) * HWN + col;
    v2f bf; bf.x = bp[0]; bf.y = bp[HWN];
#pragma unroll
    for (int t = 0; t < 4; ++t) {
      const v2f af = *(const v2f*)&sR[(t * 16 + (lane & 15)) * 66 + kk + krow];
      acc[t] = wmma_f32(af, bf, acc[t]);
    }
  }

  const int rbase = 8 * (lane >> 4);
  float* ob = outp + (size_t)b * CC * HWN + col;
#pragma unroll
  for (int t = 0; t < 4; ++t)
#pragma unroll
    for (int r = 0; r < 8; ++r) {
      const int row = t * 16 + rbase + r;
      const float v = acc[t][r] + hb[(size_t)row * HWN + col];
      ob[(size_t)row * HWN] = v * fu[row];
    }
}

// ---------------------------------------------------------------------------
extern "C" void kernel_launch(void* const* d_in, const int* in_sizes, int n_in,
                              void* d_out, int out_size, void* d_ws, size_t ws_size,
                              hipStream_t stream) {
  const float* fcn  = (const float*)d_in[0];   // [B, CIN, H, W]
  const float* hm_w = (const float*)d_in[1];   // [C, CIN]
  const float* dw_w = (const float*)d_in[2];   // [C, 1, 5, 5]
  const float* fu_w = (const float*)d_in[3];   // [C, 1, 1, 1]
  float* out = (float*)d_out;

  const size_t N = (size_t)BB * CC * HWN;      // 10,485,760
  float* A0  = (float*)d_ws;                   // unnorm -> heat1 -> pre2
  float* A1  = A0 + N;                         // dwconv output (feat_a)
  float* REL = A1 + N;                         // [B, 64, 64]

  float* final_coords = out;                   // [B, C, 2]
  float* inter_coords = out + BB * CC * 2;     // [B, C, 2]
  float* out_heat     = out + 2 * BB * CC * 2; // [B, C, H, W]

  // 1) 1x1 conv GEMM (WMMA f32, TDM-staged A-operand)
  k_gemm1<<<dim3(HWN / 128, BB), 256, 0, stream>>>(fcn, hm_w, A0);
  // 2) softmax over HW (in place) + interme DSNT coords
  k_softmax_dsnt<<<BB * CC, 256, 0, stream>>>(A0, A0, inter_coords);
  // 3) depthwise 5x5 SAME
  k_dwconv<<<(int)(N / 256), 256, 0, stream>>>(A0, dw_w, A1);
  // 4) relation = feat . feat^T  (WMMA f32, split-K + f32 atomics)
  k_zero<<<(BB * CC * CC + 255) / 256, 256, 0, stream>>>(REL, BB * CC * CC);
  k_relation<<<dim3(16, BB, 8), 32, 0, stream>>>(A1, REL);
  // 5) softmax(rowmax - relation) over last dim, in place
  k_rel_softmax<<<BB * CC, 64, 0, stream>>>(REL);
  // 6) feat_e = R.feat, + feat, * fu_w  (WMMA f32)
  k_feat_e<<<dim3(HWN / 128, BB), 256, 0, stream>>>(REL, A1, fu_w, A0);
  // 7) final softmax + heatmap output + final DSNT coords
  k_softmax_dsnt<<<BB * CC, 256, 0, stream>>>(A0, out_heat, final_coords);
}